// MimoOfdmSystemModel_84662395339201
// MI455X (gfx1250) — compile-verified
//
#include <hip/hip_runtime.h>
#include <stdint.h>

#define B_  512
#define S_  4
#define K_  912
#define R_  8
#define D_  912
#define N_  1824        // 2*K
#define SH1_ 7
#define SH2_ 150
#define BP_ITERS_ 10
#define INV_SQRT2_ 0.70710678118654752f
#define SQRT2_     1.41421356237309515f

typedef unsigned int u32x4 __attribute__((ext_vector_type(4)));
typedef int          i32x8 __attribute__((ext_vector_type(8)));
typedef int          i32x4 __attribute__((ext_vector_type(4)));

// ---------------------------------------------------------------------------
// Kernel 1: per-(b,d) LMMSE equalizer + QPSK demapper -> channel LLRs (L = -llr)
// One lane per subcarrier: register-resident 4x4 complex Cholesky solve.
// ---------------------------------------------------------------------------
__global__ __launch_bounds__(256)
void equalize_demap(const int*   __restrict__ bbits,
                    const float* __restrict__ hre,
                    const float* __restrict__ him,
                    const float* __restrict__ nre,
                    const float* __restrict__ nim,
                    const int*   __restrict__ ebno,
                    float*       __restrict__ Lch) {
  int idx = blockIdx.x * blockDim.x + threadIdx.x;
  if (idx >= B_ * D_) return;
  int b = idx / D_;
  int d = idx - b * D_;

  const size_t hbase = (size_t)idx * (R_ * S_);
#if defined(__gfx1250__)
  __builtin_prefetch(hre + hbase, 0, 1);   // global_prefetch_b8
  __builtin_prefetch(him + hbase, 0, 1);
#endif

  const float no   = exp10f(-0.1f * (float)ebno[0]);
  const float nstd = sqrtf(no * 0.5f);

  // ---- encode the two code bits (2d, 2d+1) per stream, map to QPSK --------
  float xr[S_], xi[S_];
  const int* bb = bbits + (size_t)b * (S_ * K_);
#pragma unroll
  for (int s = 0; s < S_; ++s) {
    const int* bs = bb + s * K_;
    int cbit[2];
#pragma unroll
    for (int j = 0; j < 2; ++j) {
      int n = 2 * d + j;
      if (n < K_) {
        cbit[j] = bs[n];
      } else {                       // parity p[k] = b[k]^b[k-SH1]^b[k-SH2] (mod K)
        int k  = n - K_;
        int k1 = k - SH1_; if (k1 < 0) k1 += K_;
        int k2 = k - SH2_; if (k2 < 0) k2 += K_;
        cbit[j] = bs[k] ^ bs[k1] ^ bs[k2];
      }
    }
    xr[s] = (1.0f - 2.0f * (float)(cbit[0] & 1)) * INV_SQRT2_;
    xi[s] = (1.0f - 2.0f * (float)(cbit[1] & 1)) * INV_SQRT2_;
  }

  // ---- load channel (already scaled by 1/sqrt2), 128-bit loads ------------
  float hr[R_][S_], hi[R_][S_];
  const float4* hr4 = (const float4*)(hre + hbase);
  const float4* hi4 = (const float4*)(him + hbase);
#pragma unroll
  for (int r = 0; r < R_; ++r) {
    float4 a = hr4[r], c = hi4[r];
    hr[r][0] = a.x * INV_SQRT2_; hr[r][1] = a.y * INV_SQRT2_;
    hr[r][2] = a.z * INV_SQRT2_; hr[r][3] = a.w * INV_SQRT2_;
    hi[r][0] = c.x * INV_SQRT2_; hi[r][1] = c.y * INV_SQRT2_;
    hi[r][2] = c.z * INV_SQRT2_; hi[r][3] = c.w * INV_SQRT2_;
  }

  // ---- y = H x + noise ----------------------------------------------------
  float yr[R_], yi[R_];
  const float* nr = nre + (size_t)idx * R_;
  const float* ni = nim + (size_t)idx * R_;
#pragma unroll
  for (int r = 0; r < R_; ++r) {
    float ar = 0.f, ai = 0.f;
#pragma unroll
    for (int s = 0; s < S_; ++s) {
      ar += hr[r][s] * xr[s] - hi[r][s] * xi[s];
      ai += hr[r][s] * xi[s] + hi[r][s] * xr[s];
    }
    yr[r] = ar + nr[r] * nstd;
    yi[r] = ai + ni[r] * nstd;
  }

  // ---- A = H^H H + no*I (full 4x4), b = H^H y -----------------------------
  float Ar[S_][S_], Ai[S_][S_];
#pragma unroll
  for (int s = 0; s < S_; ++s) {
#pragma unroll
    for (int t = 0; t < S_; ++t) {
      float ar = (s == t) ? no : 0.0f, ai = 0.0f;
#pragma unroll
      for (int r = 0; r < R_; ++r) {
        ar += hr[r][s] * hr[r][t] + hi[r][s] * hi[r][t];
        ai += hr[r][s] * hi[r][t] - hi[r][s] * hr[r][t];
      }
      Ar[s][t] = ar; Ai[s][t] = ai;
    }
  }
  float br[S_], bi[S_];
#pragma unroll
  for (int s = 0; s < S_; ++s) {
    float ar = 0.f, ai = 0.f;
#pragma unroll
    for (int r = 0; r < R_; ++r) {
      ar += hr[r][s] * yr[r] + hi[r][s] * yi[r];
      ai += hr[r][s] * yi[r] - hi[r][s] * yr[r];
    }
    br[s] = ar; bi[s] = ai;
  }

  // ---- complex Cholesky A = L L^H (real diagonal Ld) ----------------------
  float Lr[S_][S_], Li[S_][S_], Ld[S_];
#pragma unroll
  for (int j = 0; j < S_; ++j) {
    float djj = Ar[j][j];
#pragma unroll
    for (int k = 0; k < S_; ++k) if (k < j)
      djj -= Lr[j][k] * Lr[j][k] + Li[j][k] * Li[j][k];
    djj = sqrtf(fmaxf(djj, 1e-20f));
    Ld[j] = djj;
    float inv = 1.0f / djj;
#pragma unroll
    for (int i = 0; i < S_; ++i) if (i > j) {
      float ar = Ar[i][j], ai = Ai[i][j];
#pragma unroll
      for (int k = 0; k < S_; ++k) if (k < j) {
        ar -= Lr[i][k] * Lr[j][k] + Li[i][k] * Li[j][k];   // L[i][k]*conj(L[j][k])
        ai -= Li[i][k] * Lr[j][k] - Lr[i][k] * Li[j][k];
      }
      Lr[i][j] = ar * inv; Li[i][j] = ai * inv;
    }
    Lr[j][j] = djj; Li[j][j] = 0.0f;
  }

  // ---- solve L u = b (forward), L^H z = u (backward) ----------------------
  float ur[S_], ui[S_];
#pragma unroll
  for (int i = 0; i < S_; ++i) {
    float ar = br[i], ai = bi[i];
#pragma unroll
    for (int k = 0; k < S_; ++k) if (k < i) {
      ar -= Lr[i][k] * ur[k] - Li[i][k] * ui[k];
      ai -= Lr[i][k] * ui[k] + Li[i][k] * ur[k];
    }
    float inv = 1.0f / Ld[i];
    ur[i] = ar * inv; ui[i] = ai * inv;
  }
  float zr[S_], zi[S_];
#pragma unroll
  for (int ii = S_ - 1; ii >= 0; --ii) {
    float ar = ur[ii], ai = ui[ii];
#pragma unroll
    for (int k = 0; k < S_; ++k) if (k > ii) {
      ar -= Lr[k][ii] * zr[k] + Li[k][ii] * zi[k];          // conj(L[k][ii])*z[k]
      ai -= Lr[k][ii] * zi[k] - Li[k][ii] * zr[k];
    }
    float inv = 1.0f / Ld[ii];
    zr[ii] = ar * inv; zi[ii] = ai * inv;
  }

  // ---- diag(A^-1) from columns of M = L^-1 --------------------------------
  float dia[S_];
#pragma unroll
  for (int j = 0; j < S_; ++j) {
    float mr[S_], mi[S_];
#pragma unroll
    for (int i = 0; i < S_; ++i) { mr[i] = 0.f; mi[i] = 0.f; }
    mr[j] = 1.0f / Ld[j];
#pragma unroll
    for (int i = 0; i < S_; ++i) if (i > j) {
      float ar = 0.f, ai = 0.f;
#pragma unroll
      for (int k = 0; k < S_; ++k) if (k >= j && k < i) {
        ar += Lr[i][k] * mr[k] - Li[i][k] * mi[k];
        ai += Lr[i][k] * mi[k] + Li[i][k] * mr[k];
      }
      float inv = -1.0f / Ld[i];
      mr[i] = ar * inv; mi[i] = ai * inv;
    }
    float acc = 0.f;
#pragma unroll
    for (int i = 0; i < S_; ++i) if (i >= j)
      acc += mr[i] * mr[i] + mi[i] * mi[i];
    dia[j] = acc;
  }

  // ---- LLRs: L = -llr = (2*sqrt2/no_eff) * {Re,Im}(x_hat) -----------------
#pragma unroll
  for (int s = 0; s < S_; ++s) {
    float dgg = 1.0f - no * dia[s];              // Re diag(G H) = 1 - no*Re diag(A^-1)
    dgg = fminf(fmaxf(dgg, 1e-6f), 1.0f - 1e-6f);
    float inv_d  = 1.0f / dgg;
    float no_eff = (1.0f - dgg) * inv_d;
    float g = 2.0f * SQRT2_ / no_eff;
    float2 v;
    v.x = g * (zr[s] * inv_d);
    v.y = g * (zi[s] * inv_d);
    ((float2*)(Lch + (size_t)(b * S_ + s) * N_))[d] = v;   // positions 2d, 2d+1
  }
}

// ---------------------------------------------------------------------------
// Kernel 2: min-sum BP decode, one workgroup (8 waves) per codeword.
// LDS-resident (29 KB of 320 KB/WGP). Channel LLRs staged via TDM tensor load.
// ---------------------------------------------------------------------------
__global__ __launch_bounds__(256)
void bp_decode(const float* __restrict__ Lch,
               const int*   __restrict__ bbits,
               float*       __restrict__ out) {
  __shared__ float Lc[N_];
  __shared__ float tot[N_];
  __shared__ float c2v[K_ * 4];

  const int cw  = blockIdx.x;            // codeword = b*S + s
  const int tid = threadIdx.x;
  const float* Lg = Lch + (size_t)cw * N_;

#if defined(__gfx1250__) && __has_builtin(__builtin_amdgcn_tensor_load_to_lds) && __has_builtin(__builtin_amdgcn_s_wait_tensorcnt)
  if (tid < 32) {                        // wave 0 issues the TDM descriptor
    unsigned lds_off = (unsigned)(uintptr_t)(void*)&Lc[0];         // LDS byte offset
    unsigned long long ga = (unsigned long long)(uintptr_t)Lg;
    u32x4 g0; i32x8 g1;
    i32x4 g2 = {0, 0, 0, 0}, g3 = {0, 0, 0, 0};
    i32x8 g4 = {0, 0, 0, 0, 0, 0, 0, 0};
    // D# group0: count=1 | lds_addr | global_addr[56:0] | type=2
    g0[0] = 1u;
    g0[1] = lds_off;
    g0[2] = (unsigned)(ga & 0xFFFFFFFFull);
    g0[3] = (unsigned)((ga >> 32) & 0x1FFFFFFull) | (2u << 30);
    // D# group1: data_size=4B, tensor_dim0=N_, tensor_dim1=1,
    //            tile_dim0=N_, tile_dim1=1, tensor_dim0_stride=N_
    g1[0] = (int)(2u << 16);                 // workgroup_mask=0, data_size=2 (4B)
    g1[1] = (int)((unsigned)(N_ & 0xFFFF) << 16);            // dim0 lo16 @ [63:48]
    g1[2] = (int)((N_ >> 16) & 0xFFFF) | (int)(1u << 16);    // dim0 hi | dim1 lo16=1
    g1[3] = (int)((unsigned)(N_ & 0xFFFF) << 16);            // dim1 hi=0 | tile_dim0
    g1[4] = 1;                                               // tile_dim1=1, tile_dim2=0
    g1[5] = N_;                                              // dim0_stride lo32
    g1[6] = 0;
    g1[7] = 0;
    // 6-arg form (clang-23 / therock-10.0 headers): groups 0..3 + extra group + cpol
    __builtin_amdgcn_tensor_load_to_lds(g0, g1, g2, g3, g4, 0);
    __builtin_amdgcn_s_wait_tensorcnt(0);
  }
#endif
  __syncthreads();

  // Defensive cooperative (re)load of the same data + init tot / c2v.
  for (int v = tid; v < N_; v += 256) { float x = Lg[v]; Lc[v] = x; tot[v] = x; }
  for (int i = tid; i < K_ * 4; i += 256) c2v[i] = 0.0f;
  __syncthreads();

  for (int it = 0; it < BP_ITERS_; ++it) {
    // ---- check-node phase: check c touches {c, c-SH1, c-SH2, K+c} ---------
    for (int c = tid; c < K_; c += 256) {
      int v1 = c - SH1_; if (v1 < 0) v1 += K_;
      int v2 = c - SH2_; if (v2 < 0) v2 += K_;
      float m0 = tot[c]      - c2v[4 * c + 0];
      float m1 = tot[v1]     - c2v[4 * c + 1];
      float m2 = tot[v2]     - c2v[4 * c + 2];
      float m3 = tot[K_ + c] - c2v[4 * c + 3];
      float a0 = fabsf(m0), a1 = fabsf(m1), a2 = fabsf(m2), a3 = fabsf(m3);
      float min1 = a0, min2 = 3.4e38f; int am = 0;
      if (a1 < min1) { min2 = min1; min1 = a1; am = 1; } else if (a1 < min2) min2 = a1;
      if (a2 < min1) { min2 = min1; min1 = a2; am = 2; } else if (a2 < min2) min2 = a2;
      if (a3 < min1) { min2 = min1; min1 = a3; am = 3; } else if (a3 < min2) min2 = a3;
      float s0 = (m0 < 0.f) ? -1.f : 1.f;
      float s1 = (m1 < 0.f) ? -1.f : 1.f;
      float s2 = (m2 < 0.f) ? -1.f : 1.f;
      float s3 = (m3 < 0.f) ? -1.f : 1.f;
      float sp = s0 * s1 * s2 * s3;
      c2v[4 * c + 0] = sp * s0 * ((am == 0) ? min2 : min1);
      c2v[4 * c + 1] = sp * s1 * ((am == 1) ? min2 : min1);
      c2v[4 * c + 2] = sp * s2 * ((am == 2) ? min2 : min1);
      c2v[4 * c + 3] = sp * s3 * ((am == 3) ? min2 : min1);
    }
    __syncthreads();
    // ---- variable-node phase: var v in checks {v(p0), v+SH1(p1), v+SH2(p2)};
    //      parity var K+k in check k (p3) ----------------------------------
    for (int v = tid; v < N_; v += 256) {
      float e;
      if (v < K_) {
        int c1 = v + SH1_; if (c1 >= K_) c1 -= K_;
        int c2 = v + SH2_; if (c2 >= K_) c2 -= K_;
        e = c2v[4 * v + 0] + c2v[4 * c1 + 1] + c2v[4 * c2 + 2];
      } else {
        e = c2v[4 * (v - K_) + 3];
      }
      tot[v] = Lc[v] + e;
    }
    __syncthreads();
  }

  // ---- outputs: d_out = [b_bits (as float) | b_hat] -----------------------
  const int* bg = bbits + (size_t)cw * K_;
  float* o1 = out + (size_t)cw * K_;
  float* o2 = out + (size_t)(B_ * S_) * K_ + (size_t)cw * K_;
  for (int v = tid; v < K_; v += 256) {
    o1[v] = (float)bg[v];
    o2[v] = (tot[v] < 0.0f) ? 1.0f : 0.0f;
  }
}

// ---------------------------------------------------------------------------
extern "C" void kernel_launch(void* const* d_in, const int* in_sizes, int n_in,
                              void* d_out, int out_size, void* d_ws, size_t ws_size,
                              hipStream_t stream) {
  (void)in_sizes; (void)n_in; (void)out_size; (void)ws_size;
  const int*   bbits = (const int*)  d_in[0];
  const float* hre   = (const float*)d_in[1];
  const float* him   = (const float*)d_in[2];
  const float* nre   = (const float*)d_in[3];
  const float* nim   = (const float*)d_in[4];
  const int*   ebno  = (const int*)  d_in[5];
  float* Lch = (float*)d_ws;                       // B*S*N floats = 14.9 MB scratch

  const int total = B_ * D_;
  equalize_demap<<<(total + 255) / 256, 256, 0, stream>>>(
      bbits, hre, him, nre, nim, ebno, Lch);
  bp_decode<<<B_ * S_, 256, 0, stream>>>(Lch, bbits, (float*)d_out);
}